// KernelFusion_19026705121450
// MI455X (gfx1250) — compile-verified
//
#include <hip/hip_runtime.h>
#include <math.h>

typedef float v2f __attribute__((ext_vector_type(2)));
typedef float v8f __attribute__((ext_vector_type(8)));

#define HW    65536   // 256*256
#define NB    16
#define NHID  64
#define TDIM  512

// ---------------------------------------------------------------------------
// Kernel 1: one block, 4 waves. Each wave computes a 16(batch)x16(hid) tile of
// t = text_vec @ text_proj_w^T + text_proj_b via V_WMMA_F32_16X16X4_F32,
// then folds everything into per-batch / global constants in d_ws:
//   ws[b*8 + 0..2] = u_b   = W^T t_b          (W = z_proj_w, 64x3)
//   ws[b*8 + 3]    = t_b . z_proj_b
//   ws[b*8 + 4]    = |t_b|^2
//   G = ws+128:
//   G[0..8]=Q=W^T W, G[9..11]=p=W^T b, G[12]=|b|^2,
//   G[13..21]=M=out_w*W (3x3), G[22..24]=d=out_w*b, G[25..27]=out_b,
//   G[28]=gamma, G[29]=alpha, G[30]=c, G[31..33]=w, G[34]=1/(sum w + 1e-8)
// ---------------------------------------------------------------------------
__global__ void __launch_bounds__(128) kf_precompute(
    const float* __restrict__ tv,  const float* __restrict__ tpw,
    const float* __restrict__ tpb, const float* __restrict__ zpw,
    const float* __restrict__ zpb, const float* __restrict__ outw,
    const float* __restrict__ outb, const float* __restrict__ log_gamma,
    const float* __restrict__ alpha, const float* __restrict__ cc,
    const float* __restrict__ wv,  float* __restrict__ ws)
{
    __shared__ float t_sh[NB][NHID];

    const int tid  = threadIdx.x;
    const int wave = tid >> 5;
    const int lane = tid & 31;
    const int m    = lane & 15;           // batch row (A matrix M index)
    const int koff = (lane >> 4) * 2;     // lanes 0-15: K 0,1 ; lanes 16-31: K 2,3
    const int n    = wave * 16 + m;       // hidden column handled by this lane

    // D = A(16x512) x B(512x16) accumulated over 128 WMMA_F32_16X16X4_F32 steps
    v8f acc = {0.f, 0.f, 0.f, 0.f, 0.f, 0.f, 0.f, 0.f};
    const float* arow = tv  + m * TDIM + koff;  // A[m][k]   = text_vec[m][k]
    const float* brow = tpw + n * TDIM + koff;  // B[k][n]   = text_proj_w[n][k]
#pragma unroll 4
    for (int k0 = 0; k0 < TDIM; k0 += 4) {
        v2f a; a.x = arow[k0]; a.y = arow[k0 + 1];
        v2f b; b.x = brow[k0]; b.y = brow[k0 + 1];
        acc = __builtin_amdgcn_wmma_f32_16x16x4_f32(
                  false, a, false, b, (short)0, acc, false, false);
    }

    // D layout: VGPR r -> (M=r, N=lane) lanes 0-15 ; (M=8+r, N=lane-16) lanes 16-31
    const int rowbase = (lane < 16) ? 0 : 8;
    const float bias  = tpb[n];
#pragma unroll
    for (int r = 0; r < 8; ++r)
        t_sh[rowbase + r][n] = acc[r] + bias;
    __syncthreads();

    if (tid < NB) {
        float u0 = 0.f, u1 = 0.f, u2 = 0.f, tbdot = 0.f, tt = 0.f;
        for (int o = 0; o < NHID; ++o) {
            const float tb = t_sh[tid][o];
            u0 += zpw[o * 3 + 0] * tb;
            u1 += zpw[o * 3 + 1] * tb;
            u2 += zpw[o * 3 + 2] * tb;
            tbdot += tb * zpb[o];
            tt    += tb * tb;
        }
        float* P = ws + tid * 8;
        P[0] = u0; P[1] = u1; P[2] = u2; P[3] = tbdot; P[4] = tt;
    } else if (tid == 16) {
        float* G = ws + 128;
        for (int c0 = 0; c0 < 3; ++c0)
            for (int c1 = 0; c1 < 3; ++c1) {
                float s = 0.f;
                for (int o = 0; o < NHID; ++o) s += zpw[o * 3 + c0] * zpw[o * 3 + c1];
                G[c0 * 3 + c1] = s;
            }
        for (int c0 = 0; c0 < 3; ++c0) {
            float s = 0.f;
            for (int o = 0; o < NHID; ++o) s += zpw[o * 3 + c0] * zpb[o];
            G[9 + c0] = s;
        }
        {
            float s = 0.f;
            for (int o = 0; o < NHID; ++o) s += zpb[o] * zpb[o];
            G[12] = s;
        }
        for (int o2 = 0; o2 < 3; ++o2)
            for (int c0 = 0; c0 < 3; ++c0) {
                float s = 0.f;
                for (int o = 0; o < NHID; ++o) s += outw[o2 * NHID + o] * zpw[o * 3 + c0];
                G[13 + o2 * 3 + c0] = s;
            }
        for (int o2 = 0; o2 < 3; ++o2) {
            float s = 0.f;
            for (int o = 0; o < NHID; ++o) s += outw[o2 * NHID + o] * zpb[o];
            G[22 + o2] = s;
        }
        G[25] = outb[0]; G[26] = outb[1]; G[27] = outb[2];
        G[28] = expf(log_gamma[0]);
        G[29] = alpha[0];
        G[30] = cc[0];
        G[31] = wv[0]; G[32] = wv[1]; G[33] = wv[2];
        G[34] = 1.0f / (wv[0] + wv[1] + wv[2] + 1e-8f);
    }
}

// ---------------------------------------------------------------------------
// Kernel 2: pure streaming kernel, HBM-bound (~25 MB @ 23.3 TB/s ~ 1.1 us).
// One thread = 4 consecutive pixels; B128 loads/stores on all 3 channel planes.
// ---------------------------------------------------------------------------
__global__ void __launch_bounds__(256) kf_fused_pixels(
    const float* __restrict__ z, float* __restrict__ out,
    const float* __restrict__ ws)
{
    const int b   = blockIdx.y;
    const int idx = (blockIdx.x * 256 + threadIdx.x) * 4;

    const float* __restrict__ G = ws + 128;
    const float* __restrict__ P = ws + b * 8;

    const float Q00 = G[0],  Q01 = G[1],  Q02 = G[2];
    const float Q10 = G[3],  Q11 = G[4],  Q12 = G[5];
    const float Q20 = G[6],  Q21 = G[7],  Q22 = G[8];
    const float p0  = G[9],  p1  = G[10], p2  = G[11];
    const float bb  = G[12];
    const float M00 = G[13], M01 = G[14], M02 = G[15];
    const float M10 = G[16], M11 = G[17], M12 = G[18];
    const float M20 = G[19], M21 = G[20], M22 = G[21];
    const float d0  = G[22], d1  = G[23], d2  = G[24];
    const float ob0 = G[25], ob1 = G[26], ob2 = G[27];
    const float gamma = G[28], al = G[29], cc = G[30];
    const float w0 = G[31], w1 = G[32], w2 = G[33], winv = G[34];
    const float u0 = P[0], u1 = P[1], u2 = P[2], tbdot = P[3], tt = P[4];

    const float* zb = z + (size_t)b * 3 * HW;
    float*       ob = out + (size_t)b * 3 * HW;

    const float4 zc0 = *(const float4*)(zb + 0 * HW + idx);
    const float4 zc1 = *(const float4*)(zb + 1 * HW + idx);
    const float4 zc2 = *(const float4*)(zb + 2 * HW + idx);

    float4 o0v, o1v, o2v;
    const float* zx0 = (const float*)&zc0;
    const float* zx1 = (const float*)&zc1;
    const float* zx2 = (const float*)&zc2;
    float* ox0 = (float*)&o0v;
    float* ox1 = (float*)&o1v;
    float* ox2 = (float*)&o2v;

#pragma unroll
    for (int j = 0; j < 4; ++j) {
        const float a = zx0[j], bq = zx1[j], cq = zx2[j];

        // |z_map|^2 via 3x3 quadratic form
        const float y0  = Q00 * a + Q01 * bq + Q02 * cq;
        const float y1  = Q10 * a + Q11 * bq + Q12 * cq;
        const float y2  = Q20 * a + Q21 * bq + Q22 * cq;
        const float zQz = a * y0 + bq * y1 + cq * y2;
        const float zm2 = zQz + 2.f * (p0 * a + p1 * bq + p2 * cq) + bb;

        // z_map . t
        const float zmt = u0 * a + u1 * bq + u2 * cq + tbdot;

        const float dist  = zm2 - 2.f * zmt + tt;
        const float k_rbf = expf(-gamma * dist);
        float       k_pol = al * zmt + cc;
        k_pol *= k_pol;
        const float k = (w0 * k_rbf + w1 * zmt + w2 * k_pol) * winv;
        const float s = 1.f + 1.f / (1.f + expf(-k));

        ox0[j] = s * (M00 * a + M01 * bq + M02 * cq + d0) + ob0;
        ox1[j] = s * (M10 * a + M11 * bq + M12 * cq + d1) + ob1;
        ox2[j] = s * (M20 * a + M21 * bq + M22 * cq + d2) + ob2;
    }

    *(float4*)(ob + 0 * HW + idx) = o0v;
    *(float4*)(ob + 1 * HW + idx) = o1v;
    *(float4*)(ob + 2 * HW + idx) = o2v;
}

extern "C" void kernel_launch(void* const* d_in, const int* in_sizes, int n_in,
                              void* d_out, int out_size, void* d_ws, size_t ws_size,
                              hipStream_t stream) {
    (void)in_sizes; (void)n_in; (void)out_size; (void)ws_size;
    const float* z          = (const float*)d_in[0];
    const float* text_vec   = (const float*)d_in[1];
    const float* z_proj_w   = (const float*)d_in[2];
    const float* z_proj_b   = (const float*)d_in[3];
    const float* text_proj_w= (const float*)d_in[4];
    const float* text_proj_b= (const float*)d_in[5];
    const float* out_w      = (const float*)d_in[6];
    const float* out_b      = (const float*)d_in[7];
    const float* log_gamma  = (const float*)d_in[8];
    const float* alpha      = (const float*)d_in[9];
    const float* cc         = (const float*)d_in[10];
    const float* wv         = (const float*)d_in[11];
    float*       ws         = (float*)d_ws;
    float*       out        = (float*)d_out;

    kf_precompute<<<1, 128, 0, stream>>>(text_vec, text_proj_w, text_proj_b,
                                         z_proj_w, z_proj_b, out_w, out_b,
                                         log_gamma, alpha, cc, wv, ws);

    // 16384 float4-groups per batch plane / 256 threads = 64 blocks per batch
    kf_fused_pixels<<<dim3(HW / (4 * 256), NB), 256, 0, stream>>>(z, out, ws);
}